// TopKNormActivation_86904368268018
// MI455X (gfx1250) — compile-verified
//
#include <hip/hip_runtime.h>
#include <stdint.h>

// TopK(128)-by-|x| sparsification of rows of length 4096.
// One workgroup per row. Async global->LDS staging (CDNA5 TDM-class path),
// register-resident radix-select on abs-bits with wave32 shfl scans,
// single global read + single global write (memory-floor design).

#define D       4096
#define K       128
#define NT      256          // threads per block (8 wave32 waves)
#define EPT     (D / NT)     // 16 elements per thread
#define NW      (NT / 32)    // 8 waves
#define WH_STR  264          // privatized histogram stride (bank-rotating)

#if defined(__has_builtin)
#  if __has_builtin(__builtin_amdgcn_global_load_async_to_lds_b128)
#    define HAVE_ASYNC_LDS 1
#  endif
#  if __has_builtin(__builtin_amdgcn_s_wait_asynccnt)
#    define HAVE_WAIT_ASYNC_BUILTIN 1
#  endif
#endif

typedef int v4i __attribute__((ext_vector_type(4)));
typedef __attribute__((address_space(1))) v4i* gv4i_p;
typedef __attribute__((address_space(3))) v4i* lv4i_p;

__global__ __launch_bounds__(NT) void topk_mag_rows(const float* __restrict__ x,
                                                    float* __restrict__ out) {
  __shared__ uint32_t s_bits[D];           // staging + output transpose buffer
  __shared__ uint32_t s_wh[NW * WH_STR];   // per-wave privatized histograms
  __shared__ uint32_t s_wsum[NW];          // cross-wave scan partials
  __shared__ uint32_t s_sel[2];            // {chosen bucket, new kk}

  const int t    = threadIdx.x;
  const int w    = t >> 5;
  const int lane = t & 31;
  const size_t rowbase = (size_t)blockIdx.x * D;
  const float* __restrict__ src = x + rowbase;
  float* __restrict__ dst = out + rowbase;

  // ---------------- Stage row into LDS, coalesced 16B per lane ----------------
#if defined(HAVE_ASYNC_LDS)
  {
    gv4i_p g = (gv4i_p)(uintptr_t)src;     // global VA == AS1 address
    lv4i_p l = (lv4i_p)(void*)s_bits;      // generic -> LDS addrspace cast
#pragma unroll
    for (int j = 0; j < D / (NT * 4); ++j) {
      const int idx4 = j * NT + t;
      __builtin_amdgcn_global_load_async_to_lds_b128(g + idx4, l + idx4, 0, 0);
    }
#  if defined(HAVE_WAIT_ASYNC_BUILTIN)
    __builtin_amdgcn_s_wait_asynccnt(0);
#  else
    asm volatile("s_wait_asynccnt 0" ::: "memory");
#  endif
  }
#else
#pragma unroll
  for (int j = 0; j < D / (NT * 4); ++j) {
    const int idx4 = j * NT + t;
    uint4 v = *reinterpret_cast<const uint4*>(src + (size_t)idx4 * 4);
    *reinterpret_cast<uint4*>(&s_bits[idx4 * 4]) = v;
  }
#endif
  __syncthreads();

  // ---- Pull my contiguous chunk [t*EPT, (t+1)*EPT) into registers once ----
  uint32_t r[EPT];
  {
    const uint4* lb = reinterpret_cast<const uint4*>(&s_bits[t * EPT]);
#pragma unroll
    for (int j = 0; j < EPT / 4; ++j) {
      uint4 v = lb[j];
      r[4 * j + 0] = v.x; r[4 * j + 1] = v.y;
      r[4 * j + 2] = v.z; r[4 * j + 3] = v.w;
    }
  }

  // ---------------- Radix select: K-th largest abs-bit value ----------------
  // For non-negative IEEE floats, bit-pattern order == magnitude order.
  uint32_t prefix = 0;   // fixed high bytes of the threshold
  uint32_t kk = K;       // remaining rank (1-based, from the top)

  for (int p = 3; p >= 0; --p) {
    for (int i = t; i < NW * WH_STR; i += NT) s_wh[i] = 0u;
    __syncthreads();

    const uint32_t maskHi = (p == 3) ? 0u : (0xFFFFFFFFu << ((p + 1) * 8));
    const int sh = p * 8;
#pragma unroll
    for (int e = 0; e < EPT; ++e) {
      const uint32_t a = r[e] & 0x7FFFFFFFu;
      if ((a & maskHi) == prefix) {
        atomicAdd(&s_wh[w * WH_STR + ((a >> sh) & 0xFFu)], 1u);
      }
    }
    __syncthreads();

    // Merge privatized copies at reversed bucket (suffix == prefix of reversed),
    // then wave32 inclusive scan + cross-wave combine.
    uint32_t v = 0;
    {
      const int b = 255 - t;
#pragma unroll
      for (int ww = 0; ww < NW; ++ww) v += s_wh[ww * WH_STR + b];
    }
    uint32_t incl = v;
#pragma unroll
    for (int dlt = 1; dlt < 32; dlt <<= 1) {
      const uint32_t n = __shfl_up(incl, dlt, 32);
      if (lane >= dlt) incl += n;
    }
    if (lane == 31) s_wsum[w] = incl;
    __syncthreads();
    uint32_t off = 0;
    for (int ww = 0; ww < w; ++ww) off += s_wsum[ww];

    const uint32_t inclPrev = __shfl_up(incl, 1, 32);
    const uint32_t sfx  = off + incl;                       // suffix sum of bucket 255-t
    const uint32_t sfxN = (t == 0) ? 0u
                        : ((lane == 0) ? off : off + inclPrev);  // bucket 256-t
    if (sfx >= kk && sfxN < kk) {
      s_sel[0] = (uint32_t)(255 - t);
      s_sel[1] = kk - sfxN;        // rank within the chosen bucket
    }
    __syncthreads();
    prefix |= s_sel[0] << sh;
    kk = s_sel[1];
    __syncthreads();
  }

  const uint32_t T = prefix;  // abs-bits of the K-th largest element
  const uint32_t R = kk;      // number of ties (== T) to keep, first-index order

  // ------------- Deterministic tie ranking by index (wave scans) -------------
  uint32_t lt = 0;
#pragma unroll
  for (int e = 0; e < EPT; ++e) lt += ((r[e] & 0x7FFFFFFFu) == T) ? 1u : 0u;

  uint32_t incl = lt;
#pragma unroll
  for (int dlt = 1; dlt < 32; dlt <<= 1) {
    const uint32_t n = __shfl_up(incl, dlt, 32);
    if (lane >= dlt) incl += n;
  }
  if (lane == 31) s_wsum[w] = incl;
  __syncthreads();
  uint32_t off = 0;
  for (int ww = 0; ww < w; ++ww) off += s_wsum[ww];
  uint32_t run = off + incl - lt;   // exclusive prefix of ties before my chunk

  // ---------------- Apply threshold in registers ----------------
#pragma unroll
  for (int e = 0; e < EPT; ++e) {
    const uint32_t bits = r[e];
    const uint32_t a = bits & 0x7FFFFFFFu;
    uint32_t keep;
    if (a > T) {
      keep = 1u;
    } else if (a == T) {
      keep = (run < R) ? 1u : 0u;
      ++run;
    } else {
      keep = 0u;
    }
    r[e] = keep ? bits : 0u;
  }

  // ------- Transpose through LDS so global stores are fully coalesced -------
  {
    uint4* lb = reinterpret_cast<uint4*>(&s_bits[t * EPT]);
#pragma unroll
    for (int j = 0; j < EPT / 4; ++j) {
      uint4 v;
      v.x = r[4 * j + 0]; v.y = r[4 * j + 1];
      v.z = r[4 * j + 2]; v.w = r[4 * j + 3];
      lb[j] = v;
    }
  }
  __syncthreads();
#pragma unroll
  for (int j = 0; j < D / (NT * 4); ++j) {
    const int idx4 = j * NT + t;
    uint4 v = *reinterpret_cast<const uint4*>(&s_bits[idx4 * 4]);
    *reinterpret_cast<uint4*>(dst + (size_t)idx4 * 4) = v;
  }
}

extern "C" void kernel_launch(void* const* d_in, const int* in_sizes, int n_in,
                              void* d_out, int out_size, void* d_ws, size_t ws_size,
                              hipStream_t stream) {
  const float* x = (const float*)d_in[0];
  float* out = (float*)d_out;
  const int nrows = in_sizes[0] / D;   // 8*4096 = 32768 rows
  topk_mag_rows<<<dim3(nrows), dim3(NT), 0, stream>>>(x, out);
}